// local_NCC_3d_1760936591437
// MI455X (gfx1250) — compile-verified
//
#include <hip/hip_runtime.h>

#define HW   128
#define VOL  (HW * HW * HW)      // 2,097,152 voxels per volume

typedef __attribute__((ext_vector_type(2))) float v2f;
typedef __attribute__((ext_vector_type(8))) float v8f;

struct BoxArgs {
    const float* src[3];
    float*       dst[3];
};

#define LDS_STRIDE 35            // dword line stride: bank-conflict-free reads
#define LDS_PER_WAVE 560         // 16 lines * 35, rounded

// ---------------------------------------------------------------------------
// 9-wide 1-D box sum along one axis of a 128^3 volume, as a banded GEMM on the
// WMMA pipe:  Out(16x16) = Sum_{c=0..5}  A_c(16x4) * B_c(4x16)
//   A_c : constant 0/1 band chunk built from lane id; zero padding AND edge
//         clamping artifacts are folded into A (zero column kills that j).
//   B_c : input chunk, K rows striped across lanes, N = line index.
// One wave per 16(out-axis) x 16(line) tile; 4 line-tiles per wave reusing A.
// DCONTIG=true (filtered axis stride 1): stage the 16x32 input region through
// LDS with 4 coalesced global_load_b128, read B from LDS, store D as 2x b128.
// DCONTIG=false: lanes run along the stride-1 line axis, loads/stores are
// per-instruction coalesced already. blockIdx.z selects the channel.
// VGPR layouts per CDNA5 ISA 7.12.2; EXEC all ones per wave.
// ---------------------------------------------------------------------------
template <bool DCONTIG>
__global__ __launch_bounds__(256)
void box9_wmma_kernel(BoxArgs args, int sAxis, int sP, int sQ)
{
    __shared__ float smem[8 * LDS_PER_WAVE];   // 17.9 KB / WG (of 320 KB)

    const float* __restrict__ in  = args.src[blockIdx.z];
    float*       __restrict__ out = args.dst[blockIdx.z];

    const int lane  = threadIdx.x;          // 0..31 (one wave per threadIdx.y)
    const int row16 = lane & 15;            // A: M row | B: N col | D: N col
    const int kk0   = (lane >> 4) << 1;     // K of vector elt .x ; .y = kk0+1
    const int m0    = blockIdx.x << 4;      // output tile start on filtered axis

    // Band chunks (zero-padding folded in): a[c].{x,y} for K = kk0, kk0+1.
    v2f a[6];
#pragma unroll
    for (int c = 0; c < 6; ++c) {
        const int px = c * 4 + kk0, py = px + 1;   // relative j in [0,24)
        const int jx = m0 - 4 + px, jy = jx + 1;   // absolute axis index
        a[c].x = (px >= row16 && px <= row16 + 8 && jx >= 0 && jx < HW) ? 1.0f : 0.0f;
        a[c].y = (py >= row16 && py <= row16 + 8 && jy >= 0 && jy < HW) ? 1.0f : 0.0f;
    }

    const int mBase  = m0 + ((lane >> 4) << 3);
    const int ltBase = (blockIdx.y * blockDim.y + threadIdx.y) * 4;
    float* lds = smem + threadIdx.y * LDS_PER_WAVE;

#pragma unroll
    for (int t = 0; t < 4; ++t) {
        const int li      = (ltBase + t) * 16 + row16;
        const int lineOff = (li >> 7) * sP + (li & 127) * sQ;

        v8f acc = {};

        if constexpr (DCONTIG) {
            // ---- cooperative coalesced load: 16 lines x 32 floats @ d=m0-8
            const int dR   = m0 - 8;            // covers j in [m0-4, m0+19]
            const int sub  = lane >> 3;         // 4 lines per b128 instruction
            const int dOff = (lane & 7) << 2;   // 0,4,...,28
            const int dCl  = min(max(dR + dOff, 0), HW - 4); // valid address;
                                                // shifted slots are A-zeroed
#pragma unroll
            for (int i = 0; i < 4; ++i) {
                const int line = i * 4 + sub;
                const int gl   = (ltBase + t) * 16 + line;
                const int lOff = (gl >> 7) * sP + (gl & 127) * sQ;
                const float4 v = *(const float4*)(in + lOff + dCl);
                *(float4*)(lds + line * LDS_STRIDE + dOff) = v;
            }
            __builtin_amdgcn_wave_barrier();    // pin LDS store -> read order

#pragma unroll
            for (int c = 0; c < 6; ++c) {
                const int jr = 4 + c * 4 + kk0; // slot of absolute j = m0-4+c*4+kk0
                v2f b;
                b.x = lds[row16 * LDS_STRIDE + jr];
                b.y = lds[row16 * LDS_STRIDE + jr + 1];
                acc = __builtin_amdgcn_wmma_f32_16x16x4_f32(false, a[c], false, b,
                                                            (short)0, acc, false, false);
            }
            __builtin_amdgcn_wave_barrier();    // reads done before next t's stores

            // ---- vector store: m contiguous (sAxis==1), 2x b128 per lane
            float4 lo, hi;
            lo.x = acc[0]; lo.y = acc[1]; lo.z = acc[2]; lo.w = acc[3];
            hi.x = acc[4]; hi.y = acc[5]; hi.z = acc[6]; hi.w = acc[7];
            *(float4*)(out + lineOff + mBase)     = lo;
            *(float4*)(out + lineOff + mBase + 4) = hi;
        } else {
            // ---- lanes run along stride-1 lines: loads coalesce per instr
            float bv[12];
#pragma unroll
            for (int c = 0; c < 6; ++c) {
                const int jx = min(max(m0 - 4 + c * 4 + kk0, 0), HW - 1);
                const int jy = min(max(m0 - 3 + c * 4 + kk0, 0), HW - 1);
                bv[2 * c]     = in[lineOff + jx * sAxis];
                bv[2 * c + 1] = in[lineOff + jy * sAxis];
            }
#pragma unroll
            for (int c = 0; c < 6; ++c) {
                v2f b;
                b.x = bv[2 * c];
                b.y = bv[2 * c + 1];
                acc = __builtin_amdgcn_wmma_f32_16x16x4_f32(false, a[c], false, b,
                                                            (short)0, acc, false, false);
            }
#pragma unroll
            for (int r = 0; r < 8; ++r)         // 64B runs across lanes (sQ==1)
                out[lineOff + (mBase + r) * sAxis] = acc[r];
        }
    }
}

// ---------------------------------------------------------------------------
// Trilinear border-clamped warp: out[g][h,w,d] = sample(img[g], pos + flow)
// ---------------------------------------------------------------------------
__global__ void warp_trilinear_kernel(const float* __restrict__ img,
                                      const float* __restrict__ flo,
                                      float* __restrict__ out)
{
    const int idx = blockIdx.x * blockDim.x + threadIdx.x;   // 4*VOL threads
    const int g = idx >> 21;
    const int v = idx & (VOL - 1);
    const int d = v & 127, w = (v >> 7) & 127, h = v >> 14;

    const float* f = flo + g * 3 * VOL;
    float ch = fminf(fmaxf(f[v]           + (float)h, 0.0f), 127.0f);
    float cw = fminf(fmaxf(f[VOL + v]     + (float)w, 0.0f), 127.0f);
    float cd = fminf(fmaxf(f[2 * VOL + v] + (float)d, 0.0f), 127.0f);

    float h0f = floorf(ch), w0f = floorf(cw), d0f = floorf(cd);
    float fh = ch - h0f, fw = cw - w0f, fd = cd - d0f;
    int h0 = (int)h0f, w0 = (int)w0f, d0 = (int)d0f;
    int h1 = min(h0 + 1, 127), w1 = min(w0 + 1, 127), d1 = min(d0 + 1, 127);

    const float* im = img + g * VOL;
    auto S = [&](int hh, int ww, int dd) -> float {
        return im[(((hh << 7) + ww) << 7) + dd];
    };
    float c00 = S(h0, w0, d0) * (1.0f - fd) + S(h0, w0, d1) * fd;
    float c01 = S(h0, w1, d0) * (1.0f - fd) + S(h0, w1, d1) * fd;
    float c10 = S(h1, w0, d0) * (1.0f - fd) + S(h1, w0, d1) * fd;
    float c11 = S(h1, w1, d0) * (1.0f - fd) + S(h1, w1, d1) * fd;
    float c0  = c00 * (1.0f - fw) + c01 * fw;
    float c1  = c10 * (1.0f - fw) + c11 * fw;
    out[idx]  = c0 * (1.0f - fh) + c1 * fh;
}

// template = mean over 4 warped volumes; also emit template^2
__global__ void template_sq_kernel(const float* __restrict__ wp,
                                   float* __restrict__ t, float* __restrict__ t2)
{
    const int v = blockIdx.x * blockDim.x + threadIdx.x;
    float m = 0.25f * (wp[v] + wp[VOL + v] + wp[2 * VOL + v] + wp[3 * VOL + v]);
    t[v]  = m;
    t2[v] = m * m;
}

// per-group stat channels: J^2 and I*J in one pass
__global__ void stats_kernel(const float* __restrict__ t, const float* __restrict__ j,
                             float* __restrict__ j2, float* __restrict__ ij)
{
    const int v = blockIdx.x * blockDim.x + threadIdx.x;
    float tv = t[v], jv = j[v];
    j2[v] = jv * jv;
    ij[v] = tv * jv;
}

// cc = cross^2 / (Ivar*Jvar + eps); accumulate sum into acc[0]
__global__ void cc_reduce_kernel(const float* __restrict__ Isum,
                                 const float* __restrict__ I2sum,
                                 const float* __restrict__ Jsum,
                                 const float* __restrict__ J2sum,
                                 const float* __restrict__ IJsum,
                                 float* __restrict__ acc)
{
    __shared__ float sm[256];
    const int v = blockIdx.x * 256 + threadIdx.x;
    const float inv = 1.0f / 729.0f;
    float Is = Isum[v], I2 = I2sum[v], Js = Jsum[v], J2 = J2sum[v], IJ = IJsum[v];
    float cross = IJ - (Js * inv) * Is;
    float Ivar  = I2 - (Is * inv) * Is;
    float Jvar  = J2 - (Js * inv) * Js;
    float cc = cross * cross / (Ivar * Jvar + 1e-5f);
    sm[threadIdx.x] = cc;
    __syncthreads();
    for (int s = 128; s > 0; s >>= 1) {
        if (threadIdx.x < s) sm[threadIdx.x] += sm[threadIdx.x + s];
        __syncthreads();
    }
    if (threadIdx.x == 0) atomicAdd(acc, sm[0]);
}

// squared forward differences of flows along h/w/d -> acc[1..3]
__global__ void grad_reduce_kernel(const float* __restrict__ wp, float* __restrict__ acc)
{
    __shared__ float sm[256];
    const int idx = blockIdx.x * 256 + threadIdx.x;          // 12*VOL threads
    const int v = idx & (VOL - 1);
    const int d = v & 127, w = (v >> 7) & 127, h = v >> 14;
    const float val = wp[idx];
    float sx = 0.0f, sy = 0.0f, sz = 0.0f;
    if (h < 127) { float t = wp[idx + (HW * HW)] - val; sx = t * t; }
    if (w < 127) { float t = wp[idx + HW] - val;        sy = t * t; }
    if (d < 127) { float t = wp[idx + 1] - val;         sz = t * t; }

    float vals[3] = { sx, sy, sz };
#pragma unroll
    for (int k = 0; k < 3; ++k) {
        sm[threadIdx.x] = vals[k];
        __syncthreads();
        for (int s = 128; s > 0; s >>= 1) {
            if (threadIdx.x < s) sm[threadIdx.x] += sm[threadIdx.x + s];
            __syncthreads();
        }
        if (threadIdx.x == 0) atomicAdd(acc + 1 + k, sm[0]);
        __syncthreads();
    }
}

__global__ void finalize_kernel(const float* __restrict__ acc, float* __restrict__ out)
{
    const float nvox = 4.0f * (float)VOL;                     // cc mean denominator
    const float nd   = 12.0f * 127.0f * 128.0f * 128.0f;      // per-axis diff count
    out[0] = -(acc[0] / nvox) + (acc[1] + acc[2] + acc[3]) / (3.0f * nd);
}

// ---------------------------------------------------------------------------
extern "C" void kernel_launch(void* const* d_in, const int* in_sizes, int n_in,
                              void* d_out, int out_size, void* d_ws, size_t ws_size,
                              hipStream_t stream)
{
    (void)in_sizes; (void)n_in; (void)out_size; (void)ws_size;
    const float* images = (const float*)d_in[0];   // [1,4,1,128,128,128]
    const float* warps  = (const float*)d_in[1];   // [1,4,3,128,128,128]
    float* out = (float*)d_out;

    float* ws = (float*)d_ws;
    const size_t Vz = (size_t)VOL;
    float* Wbuf = ws;                // 4V warped; overwritten by J box-sums per group
    float* T    = ws + 4 * Vz;       // template
    float* I2s  = ws + 5 * Vz;       // I^2, then I^2 box-sum (in place)
    float* Isum = ws + 6 * Vz;       // I box-sum
    float* J2   = ws + 7 * Vz;       // per-group J^2, then its box-sum
    float* IJ   = ws + 8 * Vz;       // per-group I*J, then its box-sum
    float* tmpA = ws + 9 * Vz;       // 3x ping (per box channel)
    float* tmpB = ws + 12 * Vz;      // 3x pong (per box channel)
    float* acc  = ws + 15 * Vz;      // [ccSum, dxSum, dySum, dzSum]

    hipMemsetAsync(acc, 0, 4 * sizeof(float), stream);

    const int EB = 256;
    warp_trilinear_kernel<<<4 * VOL / EB, EB, 0, stream>>>(images, warps, Wbuf);
    template_sq_kernel<<<VOL / EB, EB, 0, stream>>>(Wbuf, T, I2s);

    const dim3 bblock(32, 8);        // 8 waves per WG, one wave per tile-chain
    // nchan channels batched per launch via blockIdx.z; per-channel ping/pong.
    auto boxBatch = [&](int nchan, const float* const* src, float* const* dst) {
        BoxArgs hArgs, wArgs, dArgs;
        for (int c = 0; c < nchan; ++c) {
            hArgs.src[c] = src[c];        hArgs.dst[c] = tmpA + c * Vz;
            wArgs.src[c] = tmpA + c * Vz; wArgs.dst[c] = tmpB + c * Vz;
            dArgs.src[c] = tmpB + c * Vz; dArgs.dst[c] = dst[c];
        }
        for (int c = nchan; c < 3; ++c) { // keep kernarg fully initialized
            hArgs.src[c] = hArgs.src[0]; hArgs.dst[c] = hArgs.dst[0];
            wArgs.src[c] = wArgs.src[0]; wArgs.dst[c] = wArgs.dst[0];
            dArgs.src[c] = dArgs.src[0]; dArgs.dst[c] = dArgs.dst[0];
        }
        const dim3 bgrid(8, 32, nchan);   // 8 m-tiles x (32*8 waves x 4 line-tiles)
        // H axis (stride 128^2), lines contiguous along D
        box9_wmma_kernel<false><<<bgrid, bblock, 0, stream>>>(hArgs, HW * HW, HW, 1);
        // W axis (stride 128), lines contiguous along D
        box9_wmma_kernel<false><<<bgrid, bblock, 0, stream>>>(wArgs, HW, HW * HW, 1);
        // D axis (stride 1): LDS-staged coalesced path
        box9_wmma_kernel<true><<<bgrid, bblock, 0, stream>>>(dArgs, 1, HW * HW, HW);
    };

    {   // I channels: T -> Isum, I^2 -> I^2sum (in place)
        const float* src[2] = { T, I2s };
        float*       dst[2] = { Isum, I2s };
        boxBatch(2, src, dst);
    }

    for (int g = 0; g < 4; ++g) {
        float* Jg = Wbuf + (size_t)g * Vz;
        stats_kernel<<<VOL / EB, EB, 0, stream>>>(T, Jg, J2, IJ);
        const float* src[3] = { Jg, J2, IJ };
        float*       dst[3] = { Jg, J2, IJ };   // J box-sum overwrites warped volume
        boxBatch(3, src, dst);
        cc_reduce_kernel<<<VOL / EB, EB, 0, stream>>>(Isum, I2s, Jg, J2, IJ, acc);
    }

    grad_reduce_kernel<<<12 * VOL / EB, EB, 0, stream>>>(warps, acc);
    finalize_kernel<<<1, 1, 0, stream>>>(acc, out);
}